// Numerical_solver_28587302322244
// MI455X (gfx1250) — compile-verified
//
#include <hip/hip_runtime.h>
#include <hip/hip_bf16.h>

typedef __attribute__((ext_vector_type(2))) float v2f;
typedef __attribute__((ext_vector_type(8))) float v8f;
typedef unsigned int u32x4 __attribute__((ext_vector_type(4)));
typedef int i32x4 __attribute__((ext_vector_type(4)));
typedef int i32x8 __attribute__((ext_vector_type(8)));

#define IMG 65536   // 256*256
#define NBATCH 16

// ---------------------------------------------------------------------------
// Twiddle matrices: Cm=cos(2*pi*k*n/256), Sm=sin(...), Sn=-sin(...).
// DFT matrix F = Cm - i*Sm (symmetric). Sn removes all runtime sign-scaling
// from the complex GEMM combinations (unit scales -> no per-WMMA v_pk_mul).
// ---------------------------------------------------------------------------
__global__ __launch_bounds__(256) void trig_kernel(float* __restrict__ Cm,
                                                   float* __restrict__ Sm,
                                                   float* __restrict__ Sn) {
    int idx = blockIdx.x * 256 + threadIdx.x;     // 0..65535
    int k = idx >> 8, n = idx & 255;
    int m = (k * n) & 255;                        // angle = pi * m/128
    float t = (float)m * (1.0f / 128.0f);
    float s = sinpif(t);
    Cm[idx] = cospif(t);
    Sm[idx] = s;
    Sn[idx] = -s;
}

// ---------------------------------------------------------------------------
// Deterministic two-stage reduction of wx (channel 0 of [B,4,H,W]).
// ---------------------------------------------------------------------------
__global__ __launch_bounds__(256) void reduce1_kernel(const float* __restrict__ X,
                                                      float* __restrict__ part) {
    __shared__ float sm[256];
    int j = blockIdx.x, t = threadIdx.x;          // 1024 blocks
    int b = j >> 6, chunk = j & 63;
    const float* p = X + (size_t)b * 4 * IMG + (size_t)chunk * 1024;
    float s = p[t] + p[t + 256] + p[t + 512] + p[t + 768];
    sm[t] = s;
    __syncthreads();
    for (int o = 128; o > 0; o >>= 1) {
        if (t < o) sm[t] += sm[t + o];
        __syncthreads();
    }
    if (t == 0) part[j] = sm[0];
}

__global__ void reduce2_kernel(const float* __restrict__ part, float* __restrict__ sumv) {
    if (threadIdx.x == 0 && blockIdx.x == 0) {
        float s = 0.f;
        for (int i = 0; i < 1024; ++i) s += part[i];
        sumv[0] = s;
    }
}

// ---------------------------------------------------------------------------
// Batched 256x256x256 GEMM on the f32 WMMA pipe:
//   D_b = sOut * ( A1_b * B1_b  [ + A2_b * B2_b ] )     (TERMS = 1 or 2)
// Per-operand batch stride (0 => shared twiddle matrix). Row-major.
// One wave = one 16x16 tile; block = 8 waves (same tM); grid = (32, 16).
// ---------------------------------------------------------------------------
template <int TERMS>
__global__ __launch_bounds__(256) void gemm256_kernel(
    const float* __restrict__ A1, unsigned long long sA1,
    const float* __restrict__ B1, unsigned long long sB1,
    const float* __restrict__ A2, unsigned long long sA2,
    const float* __restrict__ B2, unsigned long long sB2,
    float* __restrict__ D, float sOut) {
    int b = blockIdx.y;
    int wave = threadIdx.x >> 5;
    int lane = threadIdx.x & 31;
    int tile = blockIdx.x * 8 + wave;             // 0..255
    int tM = (tile >> 4) << 4;
    int tN = (tile & 15) << 4;
    int l16 = lane & 15;
    int koff = (lane >> 4) << 1;                  // 0 or 2

    const float* A1b = A1 + (size_t)b * sA1;
    const float* B1b = B1 + (size_t)b * sB1;
    const float* A2b = (TERMS > 1) ? (A2 + (size_t)b * sA2) : nullptr;
    const float* B2b = (TERMS > 1) ? (B2 + (size_t)b * sB2) : nullptr;

    v8f acc = {};
    for (int k0 = 0; k0 < 256; k0 += 4) {
        int kk = k0 + koff;
        // A fragment: lanes 0-15 -> K=kk..kk+1, lanes 16-31 -> K=kk..kk+1 (ISA 7.12.2)
        v2f a = *(const v2f*)(A1b + (size_t)(tM + l16) * 256 + kk);
        v2f bf;
        bf.x = B1b[(size_t)kk * 256 + tN + l16];
        bf.y = B1b[(size_t)(kk + 1) * 256 + tN + l16];
        acc = __builtin_amdgcn_wmma_f32_16x16x4_f32(false, a, false, bf,
                                                    (short)0, acc, false, false);
        if (TERMS > 1) {
            v2f a2 = *(const v2f*)(A2b + (size_t)(tM + l16) * 256 + kk);
            v2f b2;
            b2.x = B2b[(size_t)kk * 256 + tN + l16];
            b2.y = B2b[(size_t)(kk + 1) * 256 + tN + l16];
            acc = __builtin_amdgcn_wmma_f32_16x16x4_f32(false, a2, false, b2,
                                                        (short)0, acc, false, false);
        }
    }
    // C/D layout: VGPR j -> row tM+j (lanes 0-15) / tM+8+j (lanes 16-31)
    float* Db = D + (size_t)b * IMG;
    int rbase = tM + ((lane >> 4) << 3);
    int col = tN + l16;
#pragma unroll
    for (int j = 0; j < 8; ++j)
        Db[(size_t)(rbase + j) * 256 + col] = acc[j] * sOut;
}

// ---------------------------------------------------------------------------
// Spectral pointwise op:  G = -i*(Zx*KX + Zy*KY)/rad, DC <- sumv.
// kx[j] = 2*pi*fftfreq(256,dx)[j] = (pi/2)*fj with dx=2/128.
// ---------------------------------------------------------------------------
__global__ __launch_bounds__(256) void spectral_kernel(
    const float* __restrict__ Zrx, const float* __restrict__ Zix,
    const float* __restrict__ Zry, const float* __restrict__ Ziy,
    float* __restrict__ Gr, float* __restrict__ Gi,
    const float* __restrict__ sumv) {
    size_t idx = (size_t)blockIdx.x * 256 + threadIdx.x;  // 0..16*IMG-1
    int rc = (int)(idx & (IMG - 1));
    int r = rc >> 8, c = rc & 255;
    const float PI_2 = 1.57079632679489662f;
    float fc = (c < 128) ? (float)c : (float)(c - 256);
    float fr = (r < 128) ? (float)r : (float)(r - 256);
    float kx = fc * PI_2, ky = fr * PI_2;
    float rad = kx * kx + ky * ky;
    if (rc == 0) rad = 1.0f;
    float ar = Zrx[idx] * kx + Zry[idx] * ky;
    float ai = Zix[idx] * kx + Ziy[idx] * ky;
    float gr = ai / rad;
    float gi = -ar / rad;
    if (rc == 0) { gr = sumv[0]; gi = 0.0f; }
    Gr[idx] = gr;
    Gi[idx] = gi;
}

// ---------------------------------------------------------------------------
// 64 velocity-Verlet steps, one image fully resident in LDS (256KB <= 320KB).
// Initial u fill via the Tensor Data Mover: wave 0 issues one 2D D#
// (256x256 f32 tile, stride 256) and waits on TENSORcnt; the workgroup
// barrier releases the other 31 waves.
// 1024 threads (32 wave32s), thread t owns pixels p = t + i*1024 (i<64):
// col = t&255 is lane-consecutive -> conflict-free LDS stencils.
// lap0 kept in 64 VGPRs; ut streams through L2; c^2 recomputed from vel.
// ---------------------------------------------------------------------------
__global__ __launch_bounds__(1024) void verlet_kernel(
    const float* __restrict__ X, const float* __restrict__ Ug,
    float* __restrict__ UTg, float* __restrict__ out) {
    extern __shared__ float su[];                // 65536 floats
    int b = blockIdx.x, t = threadIdx.x;
    const float* wtc = X + ((size_t)b * 4 + 2) * IMG;
    const float* vel = X + ((size_t)b * 4 + 3) * IMG;
    const float* ug = Ug + (size_t)b * IMG;
    float* ut = UTg + (size_t)b * IMG;

#if __has_builtin(__builtin_amdgcn_tensor_load_to_lds) && __has_builtin(__builtin_amdgcn_s_wait_tensorcnt)
    if (t < 32) {                                // wave 0 only (uniform)
        unsigned long long ga = (unsigned long long)(const void*)ug;
        u32x4 g0;
        g0.x = 1u;                               // count=1, user mode
        g0.y = 0u;                               // lds_addr = base of dynamic LDS
        g0.z = (unsigned)(ga & 0xFFFFFFFFu);     // global_addr[31:0]
        g0.w = (unsigned)((ga >> 32) & 0x1FFFFFFu) | (2u << 30);  // addr[56:32], type=2
        i32x8 g1;
        g1[0] = 0x00020000;                      // data_size=4B, no mask/flags
        g1[1] = (int)(256u << 16);               // tensor_dim0[15:0]=256
        g1[2] = (int)(256u << 16);               // tensor_dim1[15:0]=256
        g1[3] = (int)(256u << 16);               // tile_dim0=256
        g1[4] = 256;                             // tile_dim1=256, tile_dim2=0
        g1[5] = 256;                             // tensor_dim0_stride=256
        g1[6] = 0;
        g1[7] = 0;
        i32x4 zz4 = {0, 0, 0, 0};                // groups 2/3 unused (2D tensor)
        i32x8 zz8 = {0, 0, 0, 0, 0, 0, 0, 0};
        __builtin_amdgcn_tensor_load_to_lds(g0, g1, zz4, zz4, zz8, 0);
        __builtin_amdgcn_s_wait_tensorcnt((short)0);
    }
#pragma unroll
    for (int i = 0; i < 64; ++i) {
        int p = t + i * 1024;
        ut[p] = wtc[p] * vel[p];                 // ut = wtc * c
    }
#else
#pragma unroll
    for (int i = 0; i < 64; ++i) {
        int p = t + i * 1024;
        su[p] = ug[p];
        ut[p] = wtc[p] * vel[p];
    }
#endif
    __syncthreads();

    const float inv_dx2 = 4096.0f;               // 1/dx^2, dx = 2/128
    const float DTc = 0.001f;
    const float HDT2 = 0.5f * DTc * DTc;
    const float HDT = 0.5f * DTc;
    int c = t & 255, r0 = t >> 8;
    int cp = (c + 1) & 255, cm = (c - 1) & 255;

    float lap0[64];
    for (int step = 0; step < 64; ++step) {
#pragma unroll
        for (int i = 0; i < 64; ++i) {
            int r = r0 + i * 4;
            int rp = (r + 1) & 255, rm = (r - 1) & 255;
            int row = r << 8;
            lap0[i] = (su[row + cp] + su[row + cm] + su[(rp << 8) + c] +
                       su[(rm << 8) + c] - 4.0f * su[row + c]) * inv_dx2;
        }
        __syncthreads();
#pragma unroll
        for (int i = 0; i < 64; ++i) {
            int p = ((r0 + i * 4) << 8) + c;
            float cc = vel[p];
            su[p] = su[p] + DTc * ut[p] + HDT2 * cc * cc * lap0[i];
        }
        __syncthreads();
#pragma unroll
        for (int i = 0; i < 64; ++i) {
            int r = r0 + i * 4;
            int rp = (r + 1) & 255, rm = (r - 1) & 255;
            int row = r << 8;
            float lap1 = (su[row + cp] + su[row + cm] + su[(rp << 8) + c] +
                          su[(rm << 8) + c] - 4.0f * su[row + c]) * inv_dx2;
            float cv = vel[row + c];
            ut[row + c] += HDT * cv * cv * (lap0[i] + lap1);
        }
        __syncthreads();                          // guard next step's su writes
    }

    const float inv_dx = 64.0f;
    float* o0 = out + ((size_t)b * 4 + 0) * IMG;
    float* o1 = out + ((size_t)b * 4 + 1) * IMG;
    float* o2 = out + ((size_t)b * 4 + 2) * IMG;
    float* o3 = out + ((size_t)b * 4 + 3) * IMG;
#pragma unroll
    for (int i = 0; i < 64; ++i) {
        int r = r0 + i * 4;
        int p = (r << 8) + c;
        float uc = su[p];
        o0[p] = (c == 0) ? 0.0f : (uc - su[p - 1]) * inv_dx;
        o1[p] = (r == 0) ? 0.0f : (uc - su[p - 256]) * inv_dx;
        o2[p] = ut[p] / vel[p];
        o3[p] = vel[p];
    }
}

// ---------------------------------------------------------------------------
extern "C" void kernel_launch(void* const* d_in, const int* in_sizes, int n_in,
                              void* d_out, int out_size, void* d_ws, size_t ws_size,
                              hipStream_t stream) {
    (void)in_sizes; (void)n_in; (void)out_size; (void)ws_size;
    const float* X = (const float*)d_in[0];      // [16,4,256,256] f32
    float* out = (float*)d_out;
    float* ws = (float*)d_ws;

    const size_t TOT = (size_t)NBATCH * IMG;     // 1M floats per slot
    float* S0 = ws + 0 * TOT;                    // Y_r  -> later G_r
    float* S1 = ws + 1 * TOT;                    // Y_i  -> later G_i
    float* S2 = ws + 2 * TOT;                    // P_r
    float* S3 = ws + 3 * TOT;                    // P_i
    float* S4 = ws + 4 * TOT;                    // Zr_x -> later u
    float* S5 = ws + 5 * TOT;                    // Zi_x -> later ut
    float* S6 = ws + 6 * TOT;                    // Zr_y
    float* S7 = ws + 7 * TOT;                    // Zi_y
    float* Cm = ws + 8 * TOT;                    // twiddles (256KB each)
    float* Sm = Cm + IMG;
    float* Sn = Sm + IMG;                        // -sin
    float* part = Sn + IMG;                      // 1024 partials
    float* sumv = part + 1024;

    trig_kernel<<<256, 256, 0, stream>>>(Cm, Sm, Sn);
    reduce1_kernel<<<1024, 256, 0, stream>>>(X, part);
    reduce2_kernel<<<1, 64, 0, stream>>>(part, sumv);

    dim3 gg(32, NBATCH);
    const unsigned long long sIn = 4ULL * IMG;   // batch stride in input tensor
    const unsigned long long sW = IMG;           // batch stride in workspace slot

    // FFT2 of wx:  Y = x*F (row pass), Z = F*Y (col pass); F = Cm - i*Sm
    gemm256_kernel<1><<<gg, 256, 0, stream>>>(X, sIn, Cm, 0,
                                              nullptr, 0, nullptr, 0, S0, 1.f);
    gemm256_kernel<1><<<gg, 256, 0, stream>>>(X, sIn, Sn, 0,
                                              nullptr, 0, nullptr, 0, S1, 1.f);
    gemm256_kernel<2><<<gg, 256, 0, stream>>>(Cm, 0, S0, sW,
                                              Sm, 0, S1, sW, S4, 1.f);  // Zr_x
    gemm256_kernel<2><<<gg, 256, 0, stream>>>(Cm, 0, S1, sW,
                                              Sn, 0, S0, sW, S5, 1.f);  // Zi_x
    // FFT2 of wy
    gemm256_kernel<1><<<gg, 256, 0, stream>>>(X + IMG, sIn, Cm, 0,
                                              nullptr, 0, nullptr, 0, S0, 1.f);
    gemm256_kernel<1><<<gg, 256, 0, stream>>>(X + IMG, sIn, Sn, 0,
                                              nullptr, 0, nullptr, 0, S1, 1.f);
    gemm256_kernel<2><<<gg, 256, 0, stream>>>(Cm, 0, S0, sW,
                                              Sm, 0, S1, sW, S6, 1.f);  // Zr_y
    gemm256_kernel<2><<<gg, 256, 0, stream>>>(Cm, 0, S1, sW,
                                              Sn, 0, S0, sW, S7, 1.f);  // Zi_y

    // hatv = -i*(Zx*KX + Zy*KY)/rad, DC <- sumv   (writes G into S0/S1)
    spectral_kernel<<<4096, 256, 0, stream>>>(S4, S5, S6, S7, S0, S1, sumv);

    // IFFT2: M = conj(F) = Cm + i*Sm;  P = G*M;  u = Re(M*P)/65536
    gemm256_kernel<2><<<gg, 256, 0, stream>>>(S0, sW, Cm, 0,
                                              S1, sW, Sn, 0, S2, 1.f);  // Pr
    gemm256_kernel<2><<<gg, 256, 0, stream>>>(S0, sW, Sm, 0,
                                              S1, sW, Cm, 0, S3, 1.f);  // Pi
    gemm256_kernel<2><<<gg, 256, 0, stream>>>(Cm, 0, S2, sW,
                                              Sn, 0, S3, sW, S4, 1.0f / 65536.0f); // u

    // 64 Verlet steps fully LDS-resident per image + energy outputs
    verlet_kernel<<<NBATCH, 1024, IMG * sizeof(float), stream>>>(X, S4, S5, out);
}